// GNN_14654428414044
// MI455X (gfx1250) — compile-verified
//
#include <hip/hip_runtime.h>

typedef __attribute__((ext_vector_type(2))) float v2f;
typedef __attribute__((ext_vector_type(8))) float v8f;

// ---------------------------------------------------------------------------
// Fused dual GEMM: out[N, CT*16] = X@W (+ scale2(row)*X2@W2) + bias, opt ReLU.
// One wave owns a 16-row strip across the FULL output width: CT accumulators,
// each A fragment (v2f per lane) is loaded once and feeds CT consecutive
// V_WMMA_F32_16X16X4_F32 ops -> 4x less A/W fetch traffic than tile-per-wave
// and CT independent WMMAs in flight per load group.
// A layout (16x4 f32): lanes 0-15 -> K+0/1 ; lanes 16-31 -> K+2/3.
// C/D: 8 VGPRs; lanes 0-15 rows M=0..7, lanes 16-31 rows M=8..15.
// ---------------------------------------------------------------------------
template <bool RELU, bool DUAL, bool SCALE2, int CT, int KDIM>
__global__ __launch_bounds__(256) void sage_gemm_wmma(
    const float* __restrict__ X, const float* __restrict__ W,
    const float* __restrict__ X2, const float* __restrict__ W2,
    const float* __restrict__ cnt, const float* __restrict__ bias,
    float* __restrict__ out, int nRows, int nRowTiles) {
  constexpr int OUTD = CT * 16;
  const int wave = (blockIdx.x * blockDim.x + threadIdx.x) >> 5;
  if (wave >= nRowTiles) return;              // wave-uniform: EXEC stays full
  const int lane = threadIdx.x & 31;
  const int row0 = wave << 4;
  const int m  = lane & 15;
  const int hi = lane >> 4;                   // 0: K+0/1, 1: K+2/3

  int rA = row0 + m;
  if (rA >= nRows) rA = nRows - 1;            // N is a multiple of 16; cheap guard

  v8f acc[CT];
#pragma unroll
  for (int ct = 0; ct < CT; ++ct) acc[ct] = (v8f){};

  {
    const float* xr = X + (size_t)rA * KDIM;
#pragma unroll
    for (int k = 0; k < KDIM; k += 4) {
      const int kk = k + 2 * hi;
      v2f a; a.x = xr[kk]; a.y = xr[kk + 1];
#pragma unroll
      for (int ct = 0; ct < CT; ++ct) {
        v2f b;
        b.x = W[(size_t)kk * OUTD + ct * 16 + m];
        b.y = W[(size_t)(kk + 1) * OUTD + ct * 16 + m];
        acc[ct] = __builtin_amdgcn_wmma_f32_16x16x4_f32(
            false, a, false, b, (short)0, acc[ct], false, false);
      }
    }
  }
  if constexpr (DUAL) {
    float scale = 1.0f;
    if constexpr (SCALE2) scale = 1.0f / fmaxf(cnt[rA], 1.0f);  // fused mean
    const float* xr = X2 + (size_t)rA * KDIM;
#pragma unroll
    for (int k = 0; k < KDIM; k += 4) {
      const int kk = k + 2 * hi;
      v2f a; a.x = xr[kk]; a.y = xr[kk + 1];
      if constexpr (SCALE2) { a.x *= scale; a.y *= scale; }
#pragma unroll
      for (int ct = 0; ct < CT; ++ct) {
        v2f b;
        b.x = W2[(size_t)kk * OUTD + ct * 16 + m];
        b.y = W2[(size_t)(kk + 1) * OUTD + ct * 16 + m];
        acc[ct] = __builtin_amdgcn_wmma_f32_16x16x4_f32(
            false, a, false, b, (short)0, acc[ct], false, false);
      }
    }
  }

  const int rbase = row0 + 8 * hi;
#pragma unroll
  for (int ct = 0; ct < CT; ++ct) {
    const int col = ct * 16 + m;
    const float bv = bias[col];
#pragma unroll
    for (int v = 0; v < 8; ++v) {
      const int r = rbase + v;
      if (r < nRows) {
        float val = acc[ct][v] + bv;
        if (RELU) val = fmaxf(val, 0.0f);
        out[(size_t)r * OUTD + col] = val;
      }
    }
  }
}

// ---------------------------------------------------------------------------
// Max-pool scatter, 64 features. One wave per edge; each lane owns 2 features.
// Messages are ReLU outputs (>= 0), so non-returning integer atomicMax on the
// IEEE bits with a 0-initialized accumulator IS segment_max with the
// isolated-node fixup folded in -> native global_atomic_max_i32 in L2.
// ---------------------------------------------------------------------------
__global__ __launch_bounds__(256) void sage_scatter_max64(
    const float* __restrict__ msg, const int* __restrict__ src,
    const int* __restrict__ dst, float* __restrict__ agg, int E) {
  const int tid = blockIdx.x * blockDim.x + threadIdx.x;
  const int e = __builtin_amdgcn_readfirstlane(tid >> 5);
  if (e >= E) return;
  const int lane = threadIdx.x & 31;
  const int s = __builtin_amdgcn_readfirstlane(src[e]);
  const int d = __builtin_amdgcn_readfirstlane(dst[e]);
  const v2f v = *(const v2f*)(msg + (size_t)s * 64 + lane * 2);
  int* a = (int*)(agg + (size_t)d * 64 + lane * 2);
  atomicMax(a,     __float_as_int(v.x));
  atomicMax(a + 1, __float_as_int(v.y));
}

// ---------------------------------------------------------------------------
// Mean scatter, 32 features. One wave per edge, one lane per feature; lane 0
// bumps the per-destination degree counter. Division is fused into the final
// GEMM's X2 load, so the raw sums are consumed directly.
// ---------------------------------------------------------------------------
__global__ __launch_bounds__(256) void sage_scatter_sum32(
    const float* __restrict__ h, const int* __restrict__ src,
    const int* __restrict__ dst, float* __restrict__ sum,
    float* __restrict__ cnt, int E) {
  const int tid = blockIdx.x * blockDim.x + threadIdx.x;
  const int e = __builtin_amdgcn_readfirstlane(tid >> 5);
  if (e >= E) return;
  const int lane = threadIdx.x & 31;
  const int s = __builtin_amdgcn_readfirstlane(src[e]);
  const int d = __builtin_amdgcn_readfirstlane(dst[e]);
  atomicAdd(sum + (size_t)d * 32 + lane, h[(size_t)s * 32 + lane]);
  if (lane == 0) atomicAdd(cnt + d, 1.0f);
}

// ---------------------------------------------------------------------------
extern "C" void kernel_launch(void* const* d_in, const int* in_sizes, int n_in,
                              void* d_out, int out_size, void* d_ws,
                              size_t ws_size, hipStream_t stream) {
  const float* x   = (const float*)d_in[0];
  const int*   ei  = (const int*)d_in[1];
  const float* W1p = (const float*)d_in[2];
  const float* b1p = (const float*)d_in[3];
  const float* W1s = (const float*)d_in[4];
  const float* W1n = (const float*)d_in[5];
  const float* b1  = (const float*)d_in[6];
  const float* W2p = (const float*)d_in[7];
  const float* b2p = (const float*)d_in[8];
  const float* W2s = (const float*)d_in[9];
  const float* W2n = (const float*)d_in[10];
  const float* b2  = (const float*)d_in[11];
  const float* W3s = (const float*)d_in[12];
  const float* W3n = (const float*)d_in[13];
  const float* b3  = (const float*)d_in[14];

  const int N = in_sizes[0] / 64;
  const int E = in_sizes[1] / 2;
  const int* src = ei;
  const int* dst = ei + E;

  // Workspace (floats): msg[N*64] agg[N*64] h1[N*64] h2[N*32] sum3[N*32] cnt[N]
  float* msg  = (float*)d_ws;
  float* agg  = msg  + (size_t)N * 64;
  float* h1   = agg  + (size_t)N * 64;
  float* h2   = h1   + (size_t)N * 64;
  float* sum3 = h2   + (size_t)N * 32;
  float* cnt  = sum3 + (size_t)N * 32;

  const int rowTiles = (N + 15) / 16;
  const int gblocks  = (rowTiles * 32 + 255) / 256;
  const int eblocks  = (E * 32 + 255) / 256;

  // ---- Layer 1: SAGE pool(64 -> 64), ReLU ----
  hipMemsetAsync(agg, 0, (size_t)N * 64 * sizeof(float), stream);
  sage_gemm_wmma<true, false, false, 4, 64><<<gblocks, 256, 0, stream>>>(
      x, W1p, x, W1p, nullptr, b1p, msg, N, rowTiles);
  sage_scatter_max64<<<eblocks, 256, 0, stream>>>(msg, src, dst, agg, E);
  sage_gemm_wmma<true, true, false, 4, 64><<<gblocks, 256, 0, stream>>>(
      x, W1s, agg, W1n, nullptr, b1, h1, N, rowTiles);

  // ---- Layer 2: SAGE pool(64 -> 32) ----
  hipMemsetAsync(agg, 0, (size_t)N * 64 * sizeof(float), stream);
  sage_gemm_wmma<true, false, false, 4, 64><<<gblocks, 256, 0, stream>>>(
      h1, W2p, h1, W2p, nullptr, b2p, msg, N, rowTiles);
  sage_scatter_max64<<<eblocks, 256, 0, stream>>>(msg, src, dst, agg, E);
  sage_gemm_wmma<false, true, false, 2, 64><<<gblocks, 256, 0, stream>>>(
      h1, W2s, agg, W2n, nullptr, b2, h2, N, rowTiles);

  // ---- Layer 3: SAGE mean(32 -> 32), mean fused into GEMM ----
  hipMemsetAsync(sum3, 0, (size_t)N * 33 * sizeof(float), stream);  // sum3+cnt
  sage_scatter_sum32<<<eblocks, 256, 0, stream>>>(h2, src, dst, sum3, cnt, E);
  sage_gemm_wmma<false, true, true, 2, 32><<<gblocks, 256, 0, stream>>>(
      h2, W3s, sum3, W3n, cnt, b3, (float*)d_out, N, rowTiles);
}